// ChamferLoss_7138235646289
// MI455X (gfx1250) — compile-verified
//
#include <hip/hip_runtime.h>
#include <hip/hip_bf16.h>

typedef float v2f __attribute__((ext_vector_type(2)));
typedef float v8f __attribute__((ext_vector_type(8)));

#define B_ 4
#define N_ 8192
#define M_ 8192
#define SUBT 4                              // 16-row subtiles per wave -> 64 rows
#define JCHUNKS 4                           // column chunks per batch
#define JT_PER (M_ / 16 / JCHUNKS)          // 128 j-tiles per wave
#define WAVES_PER_BATCH ((N_ / (16 * SUBT)) * JCHUNKS)  // 128*4 = 512

// Order-preserving float <-> uint key (monotonic under unsigned compare)
__device__ __forceinline__ unsigned f2key(float f) {
  unsigned u = __float_as_uint(f);
  return (u & 0x80000000u) ? ~u : (u | 0x80000000u);
}
__device__ __forceinline__ float key2f(unsigned k) {
  unsigned u = (k & 0x80000000u) ? (k & 0x7fffffffu) : ~k;
  return __uint_as_float(u);
}

__global__ void init_keys(unsigned* keys, int n) {
  int i = blockIdx.x * blockDim.x + threadIdx.x;
  if (i < n) keys[i] = 0xffffffffu;  // +max key
}

__global__ void chamfer_wmma(const float* __restrict__ preds,
                             const float* __restrict__ gts,
                             unsigned* __restrict__ rowkeys,   // B*N
                             unsigned* __restrict__ colkeys) { // B*M
  const int lane   = threadIdx.x & 31;
  const int waveId = (blockIdx.x * blockDim.x + threadIdx.x) >> 5;
  const int b        = waveId / WAVES_PER_BATCH;
  const int rem      = waveId % WAVES_PER_BATCH;
  const int rowGroup = rem / JCHUNKS;        // 0..127
  const int jchunk   = rem % JCHUNKS;        // 0..3
  const int rowbase  = rowGroup * (16 * SUBT);
  const int jbase    = jchunk * (16 * JT_PER);

  const int n  = lane & 15;
  const int hi = lane >> 4;

  const float* xb = preds + ((size_t)b * N_ + rowbase) * 3;
  const float* yB = gts + (size_t)b * M_ * 3;

  // A fragments: 16x4 f32, lanes 0-15 hold K=0,1; lanes 16-31 hold K=2,3.
  // A row = [-2x0, -2x1, -2x2, 1].  C accumulator = |x_row|^2 per C-layout slot.
  v2f a[SUBT];
  v8f xsqv[SUBT];
#pragma unroll
  for (int s = 0; s < SUBT; ++s) {
    int row = s * 16 + n;
    float x0 = xb[row * 3 + 0], x1 = xb[row * 3 + 1], x2 = xb[row * 3 + 2];
    float xsq = x0 * x0 + x1 * x1 + x2 * x2;
    a[s].x = hi ? (-2.0f * x2) : (-2.0f * x0);
    a[s].y = hi ? 1.0f         : (-2.0f * x1);
#pragma unroll
    for (int r = 0; r < 8; ++r)  // C slot (lane, r) is row 8*hi + r
      xsqv[s][r] = __shfl(xsq, 8 * hi + r, 32);
  }

  v8f rowacc[SUBT];
#pragma unroll
  for (int s = 0; s < SUBT; ++s)
#pragma unroll
    for (int r = 0; r < 8; ++r) rowacc[s][r] = 3.0e38f;

  for (int jt = 0; jt < JT_PER; ++jt) {
    const int jcol0 = jbase + jt * 16;
    const float* yb = yB + (size_t)jcol0 * 3;
    // B fragment: B column j = [y0, y1, y2, |y|^2]^T
    float y0 = yb[n * 3 + 0], y1 = yb[n * 3 + 1], y2 = yb[n * 3 + 2];
    float ysq = y0 * y0 + y1 * y1 + y2 * y2;
    v2f bf;
    bf.x = hi ? y2  : y0;   // K=2 : K=0
    bf.y = hi ? ysq : y1;   // K=3 : K=1

    float cmin = 3.0e38f;
#pragma unroll
    for (int s = 0; s < SUBT; ++s) {
      // D = A*B + C  ==  |x|^2 + |y|^2 - 2 x.y  (full squared distance tile)
      v8f p = __builtin_amdgcn_wmma_f32_16x16x4_f32(
          false, a[s], false, bf, (short)0, xsqv[s], false, false);
#pragma unroll
      for (int r = 0; r < 8; ++r) {
        rowacc[s][r] = fminf(rowacc[s][r], p[r]);
        cmin = fminf(cmin, p[r]);
      }
    }
    // column min over all 64 rows: combine the two 8-row halves
    cmin = fminf(cmin, __shfl_xor(cmin, 16, 32));
    if (hi == 0)
      atomicMin(&colkeys[(size_t)b * M_ + jcol0 + n], f2key(cmin));
  }

  // row mins: reduce each rowacc element across the 16 lanes of its half
#pragma unroll
  for (int s = 0; s < SUBT; ++s) {
#pragma unroll
    for (int r = 0; r < 8; ++r) {
      float v = rowacc[s][r];
      v = fminf(v, __shfl_xor(v, 1, 32));
      v = fminf(v, __shfl_xor(v, 2, 32));
      v = fminf(v, __shfl_xor(v, 4, 32));
      v = fminf(v, __shfl_xor(v, 8, 32));
      if (n == r)
        atomicMin(&rowkeys[(size_t)b * N_ + rowbase + s * 16 + 8 * hi + r],
                  f2key(v));
    }
  }
}

__global__ void reduce_keys(const unsigned* __restrict__ keys, int n,
                            float* __restrict__ out) {
  __shared__ float sm[256];
  float local = 0.0f;
  for (int i = threadIdx.x; i < n; i += blockDim.x) local += key2f(keys[i]);
  sm[threadIdx.x] = local;
  __syncthreads();
  for (int s = blockDim.x / 2; s > 0; s >>= 1) {
    if ((int)threadIdx.x < s) sm[threadIdx.x] += sm[threadIdx.x + s];
    __syncthreads();
  }
  if (threadIdx.x == 0) out[0] = sm[0];
}

extern "C" void kernel_launch(void* const* d_in, const int* in_sizes, int n_in,
                              void* d_out, int out_size, void* d_ws, size_t ws_size,
                              hipStream_t stream) {
  const float* preds = (const float*)d_in[0];
  const float* gts   = (const float*)d_in[1];
  unsigned* rowkeys = (unsigned*)d_ws;                 // B*N keys
  unsigned* colkeys = rowkeys + (size_t)B_ * N_;       // B*M keys
  float* out = (float*)d_out;

  const int totalKeys = B_ * N_ + B_ * M_;             // 65536 (256 KB ws)
  init_keys<<<(totalKeys + 255) / 256, 256, 0, stream>>>(rowkeys, totalKeys);

  const int totalWaves = B_ * WAVES_PER_BATCH;         // 2048 waves
  chamfer_wmma<<<totalWaves * 32 / 256, 256, 0, stream>>>(preds, gts, rowkeys,
                                                          colkeys);

  reduce_keys<<<1, 256, 0, stream>>>(rowkeys, totalKeys, out);
}